// PostProcess_16063177687425
// MI455X (gfx1250) — compile-verified
//
#include <hip/hip_runtime.h>

#define NVOX (256*256*256)
#define NC 13
#define MAXID 64
#define PIDMAX 67

// ws int layout: [0,64) presentRaw | [64,896) counts(64x13) | [896,960) pid_of_id
#define WS_PRESENT 0
#define WS_COUNTS  64
#define WS_PID     896

__global__ void k_zero_meta(int* __restrict__ ws) {
    int i = blockIdx.x * blockDim.x + threadIdx.x;
    if (i < 1024) ws[i] = 0;
}

// Pass 1: presence flags (over ALL voxels) + surface-weighted (inst,sem) histogram.
__global__ void k_hist(const float* __restrict__ g, const int* __restrict__ inst,
                       const int* __restrict__ sem,
                       int* __restrict__ presentRaw, int* __restrict__ counts) {
    __shared__ int sHist[MAXID * NC];
    __shared__ int sFlag[MAXID];
    const int tid = threadIdx.x;
    for (int i = tid; i < MAXID * NC; i += blockDim.x) sHist[i] = 0;
    if (tid < MAXID) sFlag[tid] = 0;
    __syncthreads();

    const int stride = gridDim.x * blockDim.x;
    const int n4 = NVOX / 4;
    const float4* g4 = (const float4*)g;
    const int4*   i4 = (const int4*)inst;
    const int4*   s4 = (const int4*)sem;
    for (int v = blockIdx.x * blockDim.x + tid; v < n4; v += stride) {
        float4 gv = g4[v];
        int4   iv = i4[v];
        int4   sv = s4[v];
        sFlag[iv.x] = 1; sFlag[iv.y] = 1; sFlag[iv.z] = 1; sFlag[iv.w] = 1;
        if (fabsf(gv.x) < 1.0f) atomicAdd(&sHist[iv.x * NC + sv.x], 1);
        if (fabsf(gv.y) < 1.0f) atomicAdd(&sHist[iv.y * NC + sv.y], 1);
        if (fabsf(gv.z) < 1.0f) atomicAdd(&sHist[iv.z * NC + sv.z], 1);
        if (fabsf(gv.w) < 1.0f) atomicAdd(&sHist[iv.w * NC + sv.w], 1);
    }
    __syncthreads();
    for (int i = tid; i < MAXID * NC; i += blockDim.x) {
        int c = sHist[i];
        if (c) atomicAdd(&counts[i], c);
    }
    if (tid < MAXID && sFlag[tid]) atomicOr(&presentRaw[tid], 1);
}

// Pass 2: 64-wide reduction -> pid_of_id, sem_of_pid (written as floats).
__global__ void k_reduce(const int* __restrict__ loc, int nloc,
                         const int* __restrict__ presentRaw,
                         const int* __restrict__ counts,
                         int* __restrict__ pidOf, float* __restrict__ semOut) {
    __shared__ int sPresent[MAXID];
    __shared__ int sPid[MAXID];
    const int tid = threadIdx.x;  // 64 threads
    bool keep = false;
    if (tid > 0) {
        for (int j = 0; j < nloc; ++j)
            if (loc[j] + 1 == tid) keep = true;
    }
    sPresent[tid] = (keep && presentRaw[tid]) ? 1 : 0;
    __syncthreads();
    if (tid == 0) {
        int r = 0;
        for (int id = 0; id < MAXID; ++id) {
            if (sPresent[id]) { ++r; sPid[id] = r + 2; }
            else              { sPid[id] = 0; }
        }
    }
    __syncthreads();
    pidOf[tid] = sPid[tid];

    // first-max argmax (jnp.argmax tie-break: smallest label)
    int best = counts[tid * NC], bl = 0;
    for (int j = 1; j < NC; ++j) {
        int c = counts[tid * NC + j];
        if (c > best) { best = c; bl = j; }
    }
    for (int i = tid; i < PIDMAX; i += MAXID) semOut[i] = 0.0f;
    __syncthreads();
    if (sPresent[tid]) semOut[sPid[tid]] = (float)bl;   // pids >= 3, distinct
    __syncthreads();
    if (tid == 0) { semOut[1] = 10.0f; semOut[2] = 11.0f; }
}

// Pass 3 (fused pano-build + NN fill):
// Each block reconstructs the pre-fill pano snapshot for its 13^3 halo tile
// directly from g/inst/sem (pure per-voxel function -> identical to a global
// snapshot), then resolves unassigned surface voxels with the fixed
// dx(outer),dy,dz in [-3,2] first-nonzero scan.
#define TB 8
#define TH (TB + 5)   // halo [-3,+2] -> 13
__global__ void __launch_bounds__(TB * TB * TB)
k_fill(const float* __restrict__ g, const int* __restrict__ inst,
       const int* __restrict__ sem, const int* __restrict__ pidOf,
       float* __restrict__ out) {
    __shared__ int  tile[TH * TH * TH];
    __shared__ char sSurf[TB * TB * TB];
    __shared__ int  sPid[MAXID];

    const int tx = threadIdx.x, ty = threadIdx.y, tz = threadIdx.z;
    const int tid = (tz * TB + ty) * TB + tx;
    if (tid < MAXID) sPid[tid] = pidOf[tid];
    __syncthreads();

    const int b0 = blockIdx.z * TB, b1 = blockIdx.y * TB, b2 = blockIdx.x * TB;

    for (int idx = tid; idx < TH * TH * TH; idx += TB * TB * TB) {
        int t0 = idx / (TH * TH);
        int r  = idx - t0 * (TH * TH);
        int t1 = r / TH;
        int t2 = r - t1 * TH;
        int g0 = b0 + t0 - 3, g1 = b1 + t1 - 3, g2 = b2 + t2 - 3;
        int  v = 0;
        char surf = 0;
        if ((unsigned)g0 < 256u && (unsigned)g1 < 256u && (unsigned)g2 < 256u) {
            int gi = (g0 * 256 + g1) * 256 + g2;
            if (fabsf(g[gi]) < 1.0f) {
                surf = 1;
                int s = sem[gi];
                if (s == 10)      v = 1;
                else if (s == 11) v = 2;
                else              v = sPid[inst[gi]];
            }
        }
        tile[idx] = v;
        // capture surface flag for the center 8^3 region
        int u0 = t0 - 3, u1 = t1 - 3, u2 = t2 - 3;
        if ((unsigned)u0 < (unsigned)TB && (unsigned)u1 < (unsigned)TB &&
            (unsigned)u2 < (unsigned)TB)
            sSurf[(u0 * TB + u1) * TB + u2] = surf;
    }
    __syncthreads();

    const int c0 = tz + 3, c1 = ty + 3, c2 = tx + 3;
    const int center = tile[(c0 * TH + c1) * TH + c2];
    int result = center;
    if (center == 0 && sSurf[(tz * TB + ty) * TB + tx]) {  // unassigned surface
        int found = 0;
        for (int d0 = -3; d0 <= 2 && !found; ++d0)
            for (int d1 = -3; d1 <= 2 && !found; ++d1) {
                const int* row = &tile[((c0 + d0) * TH + (c1 + d1)) * TH + c2];
                for (int d2 = -3; d2 <= 2; ++d2) {
                    int v = row[d2];
                    if (v != 0) { found = v; break; }
                }
            }
        result = found;
    }
    const int vi = ((b0 + tz) * 256 + (b1 + ty)) * 256 + (b2 + tx);
    out[vi] = (float)result;
}

extern "C" void kernel_launch(void* const* d_in, const int* in_sizes, int n_in,
                              void* d_out, int out_size, void* d_ws, size_t ws_size,
                              hipStream_t stream) {
    const float* g    = (const float*)d_in[0];
    const int*   inst = (const int*)d_in[1];
    const int*   sem  = (const int*)d_in[2];
    const int*   loc  = (const int*)d_in[3];
    const int nloc = in_sizes[3];

    float* out = (float*)d_out;     // [0,NVOX): pano, [NVOX,NVOX+67): sem_of_pid
    int* wsp        = (int*)d_ws;   // only 4 KB of ws used
    int* presentRaw = wsp + WS_PRESENT;
    int* counts     = wsp + WS_COUNTS;
    int* pidOf      = wsp + WS_PID;

    k_zero_meta<<<4, 256, 0, stream>>>(wsp);
    k_hist<<<2048, 256, 0, stream>>>(g, inst, sem, presentRaw, counts);
    k_reduce<<<1, MAXID, 0, stream>>>(loc, nloc, presentRaw, counts, pidOf, out + NVOX);
    dim3 fb(TB, TB, TB), fg(256 / TB, 256 / TB, 256 / TB);
    k_fill<<<fg, fb, 0, stream>>>(g, inst, sem, pidOf, out);
}